// AE_SkipLSTM_AR_18915035971748
// MI455X (gfx1250) — compile-verified
//
#include <hip/hip_runtime.h>
#include <cstdint>

typedef __attribute__((ext_vector_type(2))) float v2f;
typedef __attribute__((ext_vector_type(8))) float v8f;
typedef __attribute__((ext_vector_type(4))) unsigned int u32x4;
typedef __attribute__((ext_vector_type(4))) int i32x4;
typedef __attribute__((ext_vector_type(8))) int i32x8;

#if defined(__has_builtin)
#if __has_builtin(__builtin_amdgcn_tensor_load_to_lds) && __has_builtin(__builtin_amdgcn_s_wait_tensorcnt)
#define HAVE_TDM 1
#endif
#endif
#ifndef HAVE_TDM
#define HAVE_TDM 0
#endif

__device__ __forceinline__ v8f wmma4(v2f a, v2f b, v8f c) {
    // V_WMMA_F32_16X16X4_F32: D = A(16x4) * B(4x16) + C(16x16), fp32
    return __builtin_amdgcn_wmma_f32_16x16x4_f32(false, a, false, b, (short)0, c, false, false);
}

__device__ __forceinline__ float sigmoidf_(float x) {
    return 1.0f / (1.0f + __expf(-x));
}

#if HAVE_TDM
// Issue one 2D TDM tile load: rows x cols fp32, row pitch row_stride_elems,
// global tile start gaddr -> LDS byte offset lds_off.  D# per ISA 08 §8.
__device__ __forceinline__ void tdm_load_tile(const float* gaddr, unsigned lds_off,
                                              int rows, int cols, int row_stride_elems) {
    unsigned long long ga = (unsigned long long)(uintptr_t)gaddr;
    u32x4 g0;
    g0[0] = 1u;                                              // count=1, user desc
    g0[1] = lds_off;                                         // lds_addr (bytes)
    g0[2] = (unsigned)(ga & 0xFFFFFFFFu);                    // global_addr lo
    g0[3] = (unsigned)((ga >> 32) & 0x1FFFFFFu) | (2u << 30); // addr hi | type=2
    i32x8 g1;
    g1[0] = (2 << 16);                                       // data_size=4B, mask=0
    g1[1] = (cols & 0xFFFF) << 16;                           // tensor_dim0 lo16
    g1[2] = ((cols >> 16) & 0xFFFF) | ((rows & 0xFFFF) << 16); // td0 hi | td1 lo
    g1[3] = ((rows >> 16) & 0xFFFF) | ((cols & 0xFFFF) << 16); // td1 hi | tile_dim0
    g1[4] = (rows & 0xFFFF);                                 // tile_dim1 (tile_dim2=0)
    g1[5] = row_stride_elems;                                // tensor_dim0_stride lo32
    g1[6] = 0;
    g1[7] = 0;
    i32x4 z4 = {0, 0, 0, 0};
#if __clang_major__ >= 23
    i32x8 z8 = {0, 0, 0, 0, 0, 0, 0, 0};
    __builtin_amdgcn_tensor_load_to_lds(g0, g1, z4, z4, z8, 0);
#else
    __builtin_amdgcn_tensor_load_to_lds(g0, g1, z4, z4, 0);
#endif
}
#endif

// ---------------------------------------------------------------------------
// Generic batched GEMM: C[M,N] = A[M,K]*B[K,N] + bias[N]; K % 8 == 0.
// One wave per 16x16 tile; two independent WMMA accumulation chains.
// ---------------------------------------------------------------------------
__global__ __launch_bounds__(256)
void gemm16_kernel(const float* __restrict__ A, int lda, int rpb, long long bstride,
                   const float* __restrict__ Bm, int ldb,
                   const float* __restrict__ bias,
                   float* __restrict__ C, int ldc,
                   int Mrows, int Ncols, int K) {
    int wave = (blockIdx.x * blockDim.x + threadIdx.x) >> 5;
    int ntiles = Ncols >> 4;
    int total = (Mrows >> 4) * ntiles;
    if (wave >= total) return;                 // wave-uniform exit
    int mt = wave / ntiles, nt = wave % ntiles;
    int lane = threadIdx.x & 31, half = lane >> 4, l = lane & 15;

    int row = (mt << 4) + l;
    int col = (nt << 4) + l;
    const float* arow = A + (long long)(row / rpb) * bstride + (long long)(row % rpb) * lda;

    v8f acc0 = {}, acc1 = {};
    for (int k = 0; k < K; k += 8) {
        v2f a0 = *(const v2f*)(arow + k + 2 * half);
        v2f b0; b0.x = Bm[(size_t)(k + 2 * half) * ldb + col];
                b0.y = Bm[(size_t)(k + 2 * half + 1) * ldb + col];
        acc0 = wmma4(a0, b0, acc0);
        v2f a1 = *(const v2f*)(arow + k + 4 + 2 * half);
        v2f b1; b1.x = Bm[(size_t)(k + 4 + 2 * half) * ldb + col];
                b1.y = Bm[(size_t)(k + 5 + 2 * half) * ldb + col];
        acc1 = wmma4(a1, b1, acc1);
    }
    float bv = bias ? bias[col] : 0.0f;
#pragma unroll
    for (int v = 0; v < 8; ++v) {
        int r = (mt << 4) + v + (half << 3);
        C[(size_t)r * ldc + col] = acc0[v] + acc1[v] + bv;
    }
}

// ---------------------------------------------------------------------------
// Skip-LSTM: 4 blocks x 16 batch rows, 512 threads (16 waves).
// Per step: 5 WMMA GEMMs 16x256x256 (two acc chains each), then gates.
// ---------------------------------------------------------------------------
__global__ __launch_bounds__(512)
void skiplstm_kernel(const float* __restrict__ xW,      // (64,128,1024) bias folded
                     const float* __restrict__ rk,      // (256,1024)
                     const float* __restrict__ k2,      // (256,256)
                     const float* __restrict__ bias2,   // (256)
                     const float* __restrict__ s0p,
                     float* __restrict__ hs,            // (64,128,256)
                     float* __restrict__ scratch) {
    const int T = 128, UU = 256;
    const int b0 = blockIdx.x * 16;
    float* gates = scratch + (size_t)blockIdx.x * 40960;   // 16*1024
    float* cst   = gates + 16 * 1024;                      // 16*256
    float* rbuf  = cst   + 16 * 256;                       // 16*1024
    float* sbuf  = rbuf  + 16 * 1024;                      // 16*256

    int tid = threadIdx.x;
    for (int i = tid; i < 16 * 1024; i += 512) gates[i] = 0.0f;
    for (int i = tid; i < 16 * 256;  i += 512) cst[i]   = 0.0f;
    __threadfence_block();
    __syncthreads();

    float s0 = s0p[0];
    int wave = tid >> 5, lane = tid & 31, half = lane >> 4, l = lane & 15;
    int col = wave * 16 + l;

    for (int t = 0; t < T; ++t) {
        // ---- phase 1: WMMA GEMMs (g=0..3 recurrent, g=4 skip) ----
        for (int g = 0; g < 5; ++g) {
            v8f acc0 = {}, acc1 = {};
            if (g < 4) {
                const float* arow = gates + l * 1024 + g * 256;
                const float* Bg = rk + g * 256;
                for (int k = 0; k < 256; k += 8) {
                    v2f a0 = *(const v2f*)(arow + k + 2 * half);
                    v2f b0; b0.x = Bg[(size_t)(k + 2 * half) * 1024 + col];
                            b0.y = Bg[(size_t)(k + 2 * half + 1) * 1024 + col];
                    acc0 = wmma4(a0, b0, acc0);
                    v2f a1 = *(const v2f*)(arow + k + 4 + 2 * half);
                    v2f b1; b1.x = Bg[(size_t)(k + 4 + 2 * half) * 1024 + col];
                            b1.y = Bg[(size_t)(k + 5 + 2 * half) * 1024 + col];
                    acc1 = wmma4(a1, b1, acc1);
                }
#pragma unroll
                for (int v = 0; v < 8; ++v)
                    rbuf[(v + 8 * half) * 1024 + g * 256 + col] = acc0[v] + acc1[v];
            } else {
                if (t >= 4) {              // uniform; t<4 -> prev_h == 0
                    const float* arow = hs + (size_t)(b0 + l) * T * UU + (size_t)(t - 4) * UU;
                    for (int k = 0; k < 256; k += 8) {
                        v2f a0 = *(const v2f*)(arow + k + 2 * half);
                        v2f b0; b0.x = k2[(size_t)(k + 2 * half) * 256 + col];
                                b0.y = k2[(size_t)(k + 2 * half + 1) * 256 + col];
                        acc0 = wmma4(a0, b0, acc0);
                        v2f a1 = *(const v2f*)(arow + k + 4 + 2 * half);
                        v2f b1; b1.x = k2[(size_t)(k + 4 + 2 * half) * 256 + col];
                                b1.y = k2[(size_t)(k + 5 + 2 * half) * 256 + col];
                        acc1 = wmma4(a1, b1, acc1);
                    }
                }
#pragma unroll
                for (int v = 0; v < 8; ++v)
                    sbuf[(v + 8 * half) * 256 + col] = acc0[v] + acc1[v];
            }
        }
        __threadfence_block();
        __syncthreads();

        // ---- phase 2: elementwise gates ----
        for (int e = tid; e < 16 * 256; e += 512) {
            int bl = e >> 8, u = e & 255;
            const float* xw = xW + (size_t)(b0 + bl) * T * 1024 + (size_t)t * 1024;
            float i_ = sigmoidf_(xw[u]       + rbuf[bl * 1024 + u]);
            float f_ = sigmoidf_(xw[256 + u] + rbuf[bl * 1024 + 256 + u]);
            float cb = tanhf    (xw[512 + u] + rbuf[bl * 1024 + 512 + u]);
            float o_ = sigmoidf_(xw[768 + u] + rbuf[bl * 1024 + 768 + u]);
            float c  = f_ * cst[bl * 256 + u] + i_ * cb;
            float ht = o_ * tanhf(c);
            float hk = sigmoidf_(sbuf[bl * 256 + u] + bias2[u]);
            float h  = s0 * ht + (1.0f - s0) * hk;
            hs[(size_t)(b0 + bl) * T * UU + (size_t)t * UU + u] = h;
            float* gr = gates + bl * 1024;
            gr[u] = i_; gr[256 + u] = f_; gr[512 + u] = c; gr[768 + u] = o_;
            cst[bl * 256 + u] = c;
        }
        __threadfence_block();
        __syncthreads();
    }
}

// ---------------------------------------------------------------------------
// Dense partials: part[ks](64,4096) = A(64, ksplit K) * W tile.
// 256 blocks = 32 column blocks x 8 K-splits; W tiles (32x128 fp32) streamed
// into LDS by the Tensor Data Mover, double buffered, s_wait_tensorcnt synced.
// ---------------------------------------------------------------------------
__global__ __launch_bounds__(256)
void dense_kernel(const float* __restrict__ A,
                  const float* __restrict__ W,
                  float* __restrict__ part) {
    const int K = 32768;
    int cb = blockIdx.x & 31;
    int ks = blockIdx.x >> 5;
    int c0 = cb * 128;
    int kbase = ks * 4096;
    int tid = threadIdx.x;
    int wave = tid >> 5, lane = tid & 31, half = lane >> 4, l = lane & 15;
    int colLocal = (wave << 4) + l;
    int col = c0 + colLocal;

    __shared__ float lbuf[2][32 * 128];     // 2 x 16KB W tiles

    v8f a0 = {}, a1 = {}, a2 = {}, a3 = {};

#if HAVE_TDM
    if (wave == 0) {
        tdm_load_tile(W + (size_t)kbase * 4096 + c0,
                      (unsigned)(uintptr_t)&lbuf[0][0], 32, 128, 4096);
        __builtin_amdgcn_s_wait_tensorcnt(0);
    }
    __syncthreads();
    for (int kk = 0; kk < 4096; kk += 32) {
        int cur = (kk >> 5) & 1;
        if (wave == 0 && (kk + 32) < 4096)
            tdm_load_tile(W + (size_t)(kbase + kk + 32) * 4096 + c0,
                          (unsigned)(uintptr_t)&lbuf[cur ^ 1][0], 32, 128, 4096);
        const float* Bl = &lbuf[cur][0];
#pragma unroll
        for (int kg = 0; kg < 32; kg += 4) {
            v2f b; b.x = Bl[(kg + 2 * half) * 128 + colLocal];
                   b.y = Bl[(kg + 2 * half + 1) * 128 + colLocal];
            const float* ap = A + (size_t)l * K + kbase + kk + kg + 2 * half;
            v2f a;
            a = *(const v2f*)(ap);                  a0 = wmma4(a, b, a0);
            a = *(const v2f*)(ap + 16 * (size_t)K); a1 = wmma4(a, b, a1);
            a = *(const v2f*)(ap + 32 * (size_t)K); a2 = wmma4(a, b, a2);
            a = *(const v2f*)(ap + 48 * (size_t)K); a3 = wmma4(a, b, a3);
        }
        if (wave == 0) __builtin_amdgcn_s_wait_tensorcnt(0);
        __syncthreads();
    }
#else
    for (int kk = 0; kk < 4096; kk += 32) {
        __syncthreads();
        for (int i = tid; i < 32 * 128; i += 256) {
            int r = i >> 7, c = i & 127;
            lbuf[0][i] = W[(size_t)(kbase + kk + r) * 4096 + c0 + c];
        }
        __syncthreads();
        const float* Bl = &lbuf[0][0];
#pragma unroll
        for (int kg = 0; kg < 32; kg += 4) {
            v2f b; b.x = Bl[(kg + 2 * half) * 128 + colLocal];
                   b.y = Bl[(kg + 2 * half + 1) * 128 + colLocal];
            const float* ap = A + (size_t)l * K + kbase + kk + kg + 2 * half;
            v2f a;
            a = *(const v2f*)(ap);                  a0 = wmma4(a, b, a0);
            a = *(const v2f*)(ap + 16 * (size_t)K); a1 = wmma4(a, b, a1);
            a = *(const v2f*)(ap + 32 * (size_t)K); a2 = wmma4(a, b, a2);
            a = *(const v2f*)(ap + 48 * (size_t)K); a3 = wmma4(a, b, a3);
        }
    }
#endif

    float* pb = part + (size_t)ks * 64 * 4096;
#pragma unroll
    for (int v = 0; v < 8; ++v) {
        int r = v + (half << 3);
        pb[(size_t)(r)      * 4096 + col] = a0[v];
        pb[(size_t)(r + 16) * 4096 + col] = a1[v];
        pb[(size_t)(r + 32) * 4096 + col] = a2[v];
        pb[(size_t)(r + 48) * 4096 + col] = a3[v];
    }
}

// ---------------------------------------------------------------------------
// Finish: out = sum_ks part[ks] + dense_b + Z2  (streams X once, 128 MB)
// ---------------------------------------------------------------------------
__global__ __launch_bounds__(256)
void finish_kernel(const float* __restrict__ part, const float* __restrict__ dbias,
                   const float* __restrict__ X, const float* __restrict__ fw,
                   const float* __restrict__ fbp, float* __restrict__ out) {
    int idx = blockIdx.x * blockDim.x + threadIdx.x;
    if (idx >= 64 * 128 * 32) return;
    int n = idx & 4095, m = idx & 31, i = (idx >> 5) & 127, b = idx >> 12;
    float acc = dbias[n] + fbp[0];
#pragma unroll
    for (int ks = 0; ks < 8; ++ks) acc += part[(size_t)ks * 262144 + idx];
    const float* xp = X + (size_t)b * 129 * 4096 + (size_t)(1 + i) * 4096 + m;
#pragma unroll 4
    for (int s = 0; s < 128; ++s) acc += xp[s * 32] * fw[s];
    out[idx] = acc;
}

// ---------------------------------------------------------------------------
extern "C" void kernel_launch(void* const* d_in, const int* in_sizes, int n_in,
                              void* d_out, int out_size, void* d_ws, size_t ws_size,
                              hipStream_t stream) {
    const float* X       = (const float*)d_in[0];
    const float* e_k     = (const float*)d_in[1];
    const float* e_rk    = (const float*)d_in[2];
    const float* e_k2    = (const float*)d_in[3];
    const float* e_b     = (const float*)d_in[4];
    const float* e_b2    = (const float*)d_in[5];
    const float* e_s0    = (const float*)d_in[6];
    const float* d_k     = (const float*)d_in[7];
    const float* d_rk    = (const float*)d_in[8];
    const float* d_k2    = (const float*)d_in[9];
    const float* d_b     = (const float*)d_in[10];
    const float* d_b2    = (const float*)d_in[11];
    const float* d_s0    = (const float*)d_in[12];
    const float* dense_w = (const float*)d_in[13];
    const float* dense_b = (const float*)d_in[14];
    const float* final_w = (const float*)d_in[15];
    const float* final_b = (const float*)d_in[16];
    float* out = (float*)d_out;

    float* ws      = (float*)d_ws;
    float* xw_buf  = ws;                       // 8192*1024
    float* se_buf  = xw_buf + 8388608;         // 64*128*256
    float* sd_buf  = se_buf + 2097152;         // 64*128*256
    float* st_buf  = sd_buf + 2097152;         // 4*40960
    float* part    = st_buf + 163840;          // 8*64*4096

    // 1) encoder xW (8192x32 @ 32x1024)
    gemm16_kernel<<<4096, 256, 0, stream>>>(X, 32, 128, (long long)129 * 128 * 32,
                                            e_k, 1024, e_b, xw_buf, 1024,
                                            8192, 1024, 32);
    // 2) encoder skip-LSTM
    skiplstm_kernel<<<4, 512, 0, stream>>>(xw_buf, e_rk, e_k2, e_b2, e_s0, se_buf, st_buf);
    // 3) decoder xW (8192x256 @ 256x1024)
    gemm16_kernel<<<4096, 256, 0, stream>>>(se_buf, 256, 128, (long long)128 * 256,
                                            d_k, 1024, d_b, xw_buf, 1024,
                                            8192, 1024, 256);
    // 4) decoder skip-LSTM
    skiplstm_kernel<<<4, 512, 0, stream>>>(xw_buf, d_rk, d_k2, d_b2, d_s0, sd_buf, st_buf);
    // 5) dense K-split partials (TDM-fed)
    dense_kernel<<<256, 256, 0, stream>>>(sd_buf, dense_w, part);
    // 6) reduce partials + bias + Z2
    finish_kernel<<<1024, 256, 0, stream>>>(part, dense_b, X, final_w, final_b, out);
}